// PointNetFPModule_34754875359389
// MI455X (gfx1250) — compile-verified
//
#include <hip/hip_runtime.h>

typedef __attribute__((ext_vector_type(2))) float v2f;
typedef __attribute__((ext_vector_type(8))) float v8f;
typedef int v4i32 __attribute__((vector_size(16)));

#define GLOBAL_AS __attribute__((address_space(1)))
#define LDS_AS    __attribute__((address_space(3)))

typedef GLOBAL_AS v4i32* gv4p;
typedef LDS_AS    v4i32* lv4p;

#if __has_builtin(__builtin_amdgcn_global_load_async_to_lds_b128)
#define HAVE_ASYNC_LDS 1
#else
#define HAVE_ASYNC_LDS 0
#endif

namespace {
constexpr int B_ = 8;
constexpr int N_ = 8192;     // points
constexpr int M_ = 2048;     // centers
constexpr int C_ = 256;      // center feature channels (= points feature channels)
constexpr int CIN = 512;     // concat channels
constexpr int O_ = 256;      // conv out channels (both layers)
constexpr int J_ = B_ * N_;  // 65536 columns
constexpr int TJ = 128;      // j columns per block (strip)
constexpr int KC = 64;       // k rows per LDS chunk
constexpr int LSTRIDE = TJ + 8;  // 136: bank-conflict-free row stride (16B aligned)
constexpr float EPS_NN = 1e-8f;
constexpr float EPS_BN = 1e-5f;
}

// ---- async copy of 4 floats global -> LDS (falls back to sync copy) --------
__device__ __forceinline__ void cp4_g2l(const float* g, float* l) {
#if HAVE_ASYNC_LDS
    gv4p gp = (gv4p)(unsigned long long)(uintptr_t)g;       // AS(1), 64-bit addr
    lv4p lp = (lv4p)(unsigned int)(uintptr_t)l;             // AS(3), addr[31:0]
    __builtin_amdgcn_global_load_async_to_lds_b128(gp, lp, 0, 0);
#else
    *(float4*)l = *(const float4*)g;
#endif
}

template <int N>
__device__ __forceinline__ void wait_async() {
#if HAVE_ASYNC_LDS
#if __has_builtin(__builtin_amdgcn_s_wait_asynccnt)
    __builtin_amdgcn_s_wait_asynccnt((unsigned short)N);
#else
    asm volatile("s_wait_asynccnt %0" :: "n"(N) : "memory");
#endif
#endif
}

// ---------------------------------------------------------------- zero stats
__global__ void zero_stats(float* __restrict__ s) {
    for (int i = threadIdx.x; i < 2048; i += 256) s[i] = 0.0f;
}

// ------------------------------------------------- 3-NN + interpolate + coords
__global__ __launch_bounds__(256) void knn_interp(
    const float* __restrict__ pc,     // [B][3][N]
    const float* __restrict__ cc,     // [B][3][M]
    const float* __restrict__ cf,     // [B][C][M]
    float* __restrict__ xinterp,      // [C][J]  (J = b*N + n)
    float* __restrict__ coords_out)   // [B][3][N] passthrough output
{
    __shared__ float sx[M_], sy[M_], sz[M_];
    const int b   = blockIdx.y;
    const int tid = threadIdx.x;

    const float* ccb = cc + (size_t)b * 3 * M_;
    for (int i = tid; i < M_; i += 256) {
        sx[i] = ccb[i];
        sy[i] = ccb[M_ + i];
        sz[i] = ccb[2 * M_ + i];
    }
    __syncthreads();

    const int n = blockIdx.x * 256 + tid;
    const size_t pbase = (size_t)b * 3 * N_ + n;
    const float px = pc[pbase];
    const float py = pc[pbase + N_];
    const float pz = pc[pbase + 2 * N_];
    coords_out[pbase]          = px;
    coords_out[pbase + N_]     = py;
    coords_out[pbase + 2 * N_] = pz;

    float d0 = 3.4e38f, d1 = 3.4e38f, d2 = 3.4e38f;
    int   i0 = 0, i1 = 0, i2 = 0;
    for (int mm = 0; mm < M_; ++mm) {
        const float dx = px - sx[mm];
        const float dy = py - sy[mm];
        const float dz = pz - sz[mm];
        const float d = dx * dx + dy * dy + dz * dz;
        if (d < d0)      { d2 = d1; i2 = i1; d1 = d0; i1 = i0; d0 = d; i0 = mm; }
        else if (d < d1) { d2 = d1; i2 = i1; d1 = d;  i1 = mm; }
        else if (d < d2) { d2 = d;  i2 = mm; }
    }
    float w0 = 1.0f / (d0 + EPS_NN);
    float w1 = 1.0f / (d1 + EPS_NN);
    float w2 = 1.0f / (d2 + EPS_NN);
    const float inv = 1.0f / (w0 + w1 + w2);
    w0 *= inv; w1 *= inv; w2 *= inv;

    const float* cfb = cf + (size_t)b * C_ * M_;
    const size_t j = (size_t)b * N_ + n;
    for (int c = 0; c < C_; ++c) {
        const float* row = cfb + (size_t)c * M_;
        xinterp[(size_t)c * J_ + j] = w0 * row[i0] + w1 * row[i1] + w2 * row[i2];
    }
}

// ----------------------------------------------------------------- GEMM 1
// y1[o][j] = sum_c w1[o][c] * x[c][j];  x = [interp(0..255); points_features]
// Block = 512 threads = 16 waves = all 16 o-tiles of one 128-wide j-strip.
// B tile staged through LDS (double-buffered, async DMA); A fragments
// preloaded per chunk into registers (clause of 16 b64 loads).
__global__ __launch_bounds__(512) void gemm1(
    const float* __restrict__ xinterp,  // [256][J]
    const float* __restrict__ pf,       // [B][256][N]
    const float* __restrict__ w1,       // [256][512]
    float* __restrict__ y1,             // [256][J]
    float* __restrict__ sum1,
    float* __restrict__ sq1)
{
    __shared__ float sB[2][KC * LSTRIDE];   // 2 x ~34.8 KB

    const int tid  = threadIdx.x;
    const int lane = tid & 31;
    const int wid  = tid >> 5;          // 0..15 -> o-tile
    const int o0   = wid * 16;
    const int j0   = blockIdx.x * TJ;
    const int mrow = lane & 15;
    const int kq   = lane >> 4;
    const int jn   = lane & 15;

    const int eBase = tid * 4;          // 4 slots of 4 floats per chunk
    const float* pfb = pf + (size_t)(j0 >> 13) * C_ * N_ + (j0 & (N_ - 1));

    auto stage = [&](int kbase, int buf) {
#pragma unroll
        for (int s = 0; s < 4; ++s) {
            const int e = eBase + s * 2048;
            const int r = e >> 7;       // row 0..63
            const int cc_ = e & 127;    // col
            const float* src;
            if (kbase < C_) src = xinterp + (size_t)(kbase + r) * J_ + j0 + cc_;
            else            src = pfb + (size_t)(kbase - C_ + r) * N_ + cc_;
            cp4_g2l(src, &sB[buf][r * LSTRIDE + cc_]);
        }
    };

    v8f acc[8] = {};
    const float* wr = w1 + (size_t)(o0 + mrow) * CIN + 2 * kq;

    auto compute = [&](int kbase, int buf) {
        v2f areg[16];
#pragma unroll
        for (int t = 0; t < 16; ++t) areg[t] = *(const v2f*)(wr + kbase + 4 * t);
        const float* bp = sB[buf];
#pragma unroll
        for (int t = 0; t < 16; ++t) {
            const float* x0 = bp + (4 * t + 2 * kq) * LSTRIDE + jn;
            const float* x1 = x0 + LSTRIDE;
#pragma unroll
            for (int u = 0; u < 8; ++u) {
                v2f bb;
                bb.x = x0[16 * u];
                bb.y = x1[16 * u];
                acc[u] = __builtin_amdgcn_wmma_f32_16x16x4_f32(
                    false, areg[t], false, bb, (short)0, acc[u], false, false);
            }
        }
    };

    constexpr int NCH = CIN / KC;   // 8 chunks
    stage(0, 0);
#pragma unroll 1
    for (int c = 0; c < NCH; ++c) {
        if (c + 1 < NCH) { stage((c + 1) * KC, (c + 1) & 1); wait_async<4>(); }
        else             { wait_async<0>(); }
        __syncthreads();
        compute(c * KC, c & 1);
        __syncthreads();
    }

    // epilogue: store y1, accumulate BN batch statistics
#pragma unroll
    for (int r = 0; r < 8; ++r) {
        const int row = r + 8 * kq;
        float* p = y1 + (size_t)(o0 + row) * J_ + j0 + jn;
        float s = 0.0f, q = 0.0f;
#pragma unroll
        for (int u = 0; u < 8; ++u) {
            const float v = acc[u][r];
            p[16 * u] = v;
            s += v;
            q += v * v;
        }
#pragma unroll
        for (int msk = 1; msk < 16; msk <<= 1) {
            s += __shfl_xor(s, msk, 32);
            q += __shfl_xor(q, msk, 32);
        }
        if (jn == 0) {
            atomicAdd(&sum1[o0 + row], s);
            atomicAdd(&sq1[o0 + row], q);
        }
    }
}

// --------------------------------------------------- BN scale/shift from stats
__global__ void bn_stats(const float* __restrict__ sum, const float* __restrict__ sq,
                         const float* __restrict__ gamma, const float* __restrict__ beta,
                         float* __restrict__ scale, float* __restrict__ shift)
{
    const int c = threadIdx.x;
    const float inv = 1.0f / (float)J_;
    const float mean = sum[c] * inv;
    const float var  = fmaxf(sq[c] * inv - mean * mean, 0.0f);
    const float s = gamma[c] * rsqrtf(var + EPS_BN);
    scale[c] = s;
    shift[c] = beta[c] - mean * s;
}

// ----------------------------------------------------------------- GEMM 2
// y2[o][j] = sum_c w2[o][c] * relu(scale1[c]*y1[c][j] + shift1[c])
__global__ __launch_bounds__(512) void gemm2(
    const float* __restrict__ y1,       // [256][J]
    const float* __restrict__ w2,       // [256][256]
    const float* __restrict__ scale1,
    const float* __restrict__ shift1,
    float* __restrict__ y2,             // [256][J]
    float* __restrict__ sum2,
    float* __restrict__ sq2)
{
    __shared__ float sB[2][KC * LSTRIDE];
    __shared__ float ss[O_], sh[O_];

    const int tid = threadIdx.x;
    for (int i = tid; i < O_; i += 512) { ss[i] = scale1[i]; sh[i] = shift1[i]; }

    const int lane = tid & 31;
    const int wid  = tid >> 5;
    const int o0   = wid * 16;
    const int j0   = blockIdx.x * TJ;
    const int mrow = lane & 15;
    const int kq   = lane >> 4;
    const int jn   = lane & 15;

    const int eBase = tid * 4;

    auto stage = [&](int kbase, int buf) {
#pragma unroll
        for (int s = 0; s < 4; ++s) {
            const int e = eBase + s * 2048;
            const int r = e >> 7;
            const int cc_ = e & 127;
            cp4_g2l(y1 + (size_t)(kbase + r) * J_ + j0 + cc_,
                    &sB[buf][r * LSTRIDE + cc_]);
        }
    };

    v8f acc[8] = {};
    const float* wr = w2 + (size_t)(o0 + mrow) * O_ + 2 * kq;

    auto compute = [&](int kbase, int buf) {
        v2f areg[16];
#pragma unroll
        for (int t = 0; t < 16; ++t) areg[t] = *(const v2f*)(wr + kbase + 4 * t);
        const float* bp = sB[buf];
#pragma unroll
        for (int t = 0; t < 16; ++t) {
            const int k = kbase + 4 * t + 2 * kq;
            const float* x0 = bp + (4 * t + 2 * kq) * LSTRIDE + jn;
            const float* x1 = x0 + LSTRIDE;
            const float s0 = ss[k],     h0 = sh[k];
            const float s1 = ss[k + 1], h1 = sh[k + 1];
#pragma unroll
            for (int u = 0; u < 8; ++u) {
                v2f bb;
                bb.x = fmaxf(fmaf(x0[16 * u], s0, h0), 0.0f);
                bb.y = fmaxf(fmaf(x1[16 * u], s1, h1), 0.0f);
                acc[u] = __builtin_amdgcn_wmma_f32_16x16x4_f32(
                    false, areg[t], false, bb, (short)0, acc[u], false, false);
            }
        }
    };

    constexpr int NCH = O_ / KC;    // 4 chunks
    stage(0, 0);
#pragma unroll 1
    for (int c = 0; c < NCH; ++c) {
        if (c + 1 < NCH) { stage((c + 1) * KC, (c + 1) & 1); wait_async<4>(); }
        else             { wait_async<0>(); }
        __syncthreads();
        compute(c * KC, c & 1);
        __syncthreads();
    }

#pragma unroll
    for (int r = 0; r < 8; ++r) {
        const int row = r + 8 * kq;
        float* p = y2 + (size_t)(o0 + row) * J_ + j0 + jn;
        float s = 0.0f, q = 0.0f;
#pragma unroll
        for (int u = 0; u < 8; ++u) {
            const float v = acc[u][r];
            p[16 * u] = v;
            s += v;
            q += v * v;
        }
#pragma unroll
        for (int msk = 1; msk < 16; msk <<= 1) {
            s += __shfl_xor(s, msk, 32);
            q += __shfl_xor(q, msk, 32);
        }
        if (jn == 0) {
            atomicAdd(&sum2[o0 + row], s);
            atomicAdd(&sq2[o0 + row], q);
        }
    }
}

// --------------------------------------------- final BN + ReLU, [o][j]->[B][o][n]
__global__ __launch_bounds__(256) void bn_relu_out(
    const float* __restrict__ y2,       // [256][J]
    const float* __restrict__ scale2,
    const float* __restrict__ shift2,
    float* __restrict__ out)            // [B][256][N]
{
    const size_t idx = (size_t)blockIdx.x * 256 + threadIdx.x;
    const int o = (int)(idx >> 16);
    const int j = (int)(idx & 65535);
    const float v = fmaxf(fmaf(y2[idx], scale2[o], shift2[o]), 0.0f);
    const int b = j >> 13;
    const int n = j & (N_ - 1);
    out[((size_t)b * O_ + o) * N_ + n] = v;
}

// ---------------------------------------------------------------------- launch
extern "C" void kernel_launch(void* const* d_in, const int* in_sizes, int n_in,
                              void* d_out, int out_size, void* d_ws, size_t ws_size,
                              hipStream_t stream) {
    const float* pc  = (const float*)d_in[0];   // points_coords   [8,3,8192]
    const float* cc  = (const float*)d_in[1];   // centers_coords  [8,3,2048]
    const float* cf  = (const float*)d_in[2];   // centers_features[8,256,2048]
    const float* pf  = (const float*)d_in[3];   // points_features [8,256,8192]
    const float* w1  = (const float*)d_in[4];   // [256,512]
    // d_in[5] = b1 (zeros; conv bias cancels in training-mode BN -> omitted)
    const float* g1  = (const float*)d_in[6];
    const float* be1 = (const float*)d_in[7];
    const float* w2  = (const float*)d_in[8];   // [256,256]
    // d_in[9] = b2 (same cancellation)
    const float* g2  = (const float*)d_in[10];
    const float* be2 = (const float*)d_in[11];

    float* ws   = (float*)d_ws;
    float* bufA = ws;                          // interp [256][J], reused as y2
    float* bufB = ws + (size_t)O_ * J_;        // y1     [256][J]
    float* st   = bufB + (size_t)O_ * J_;      // stats: 8 x 256 floats
    float* sum1 = st;        float* sq1 = st + 256;
    float* sc1  = st + 512;  float* sh1 = st + 768;
    float* sum2 = st + 1024; float* sq2 = st + 1280;
    float* sc2  = st + 1536; float* sh2 = st + 1792;

    float* outX      = (float*)d_out;                      // [8,256,8192]
    float* outCoords = outX + (size_t)B_ * O_ * N_;        // [8,3,8192]

    zero_stats<<<1, 256, 0, stream>>>(st);
    knn_interp<<<dim3(N_ / 256, B_), 256, 0, stream>>>(pc, cc, cf, bufA, outCoords);
    const int gemmBlocks = J_ / TJ;                        // 512 blocks x 512 threads
    gemm1<<<gemmBlocks, 512, 0, stream>>>(bufA, pf, w1, bufB, sum1, sq1);
    bn_stats<<<1, 256, 0, stream>>>(sum1, sq1, g1, be1, sc1, sh1);
    gemm2<<<gemmBlocks, 512, 0, stream>>>(bufB, w2, sc1, sh1, bufA, sum2, sq2);
    bn_stats<<<1, 256, 0, stream>>>(sum2, sq2, g2, be2, sc2, sh2);
    bn_relu_out<<<(O_ * J_) / 256, 256, 0, stream>>>(bufA, sc2, sh2, outX);
}